// GCNConv_78718160601832
// MI455X (gfx1250) — compile-verified
//
#include <hip/hip_runtime.h>
#include <hip/hip_bf16.h>

#define D_DIM 128

typedef float v2f __attribute__((ext_vector_type(2)));
typedef float v8f __attribute__((ext_vector_type(8)));

// ---------------------------------------------------------------------------
// agg = x (fold the self-connection into the accumulator init); deg = 0
// ---------------------------------------------------------------------------
__global__ __launch_bounds__(256) void gcn_init(const float* __restrict__ x,
                                                float* __restrict__ agg,
                                                float* __restrict__ deg,
                                                int n4, int nNodes) {
    int i = blockIdx.x * 256 + threadIdx.x;
    if (i < n4) {
        ((float4*)agg)[i] = ((const float4*)x)[i];
    }
    if (i < nNodes) deg[i] = 0.0f;
}

// ---------------------------------------------------------------------------
// deg[row[e]] += 1  (float atomics; exact for counts << 2^24)
// ---------------------------------------------------------------------------
__global__ __launch_bounds__(256) void gcn_degree(const int* __restrict__ row,
                                                  float* __restrict__ deg,
                                                  int nEdges) {
    int e = blockIdx.x * 256 + threadIdx.x;
    if (e < nEdges) atomicAdd(&deg[row[e]], 1.0f);
}

// ---------------------------------------------------------------------------
// deg[i] = deg[i] > 0 ? deg[i]^-0.5 : 0
// ---------------------------------------------------------------------------
__global__ __launch_bounds__(256) void gcn_rsqrt(float* __restrict__ deg, int n) {
    int i = blockIdx.x * 256 + threadIdx.x;
    if (i < n) {
        float d = deg[i];
        deg[i] = (d > 0.0f) ? rsqrtf(d) : 0.0f;
    }
}

// ---------------------------------------------------------------------------
// One wave32 per edge: lane j handles float4 slice j (32*4 = 128 features).
// agg[row] += x[col] * (dis[row]*dis[col])   via global_atomic_add_f32.
// x and agg are both L2-resident (51.2 MB each < 192 MB), so this is an
// L2-rate scatter, not an HBM one.
// ---------------------------------------------------------------------------
__global__ __launch_bounds__(256) void gcn_scatter(const float* __restrict__ x,
                                                   const int* __restrict__ row,
                                                   const int* __restrict__ col,
                                                   const float* __restrict__ dis,
                                                   float* __restrict__ agg,
                                                   int nEdges) {
    int e = blockIdx.x * 8 + (threadIdx.x >> 5);
    if (e >= nEdges) return;
    int lane = threadIdx.x & 31;
    int r = row[e];
    int c = col[e];
    float norm = dis[r] * dis[c];
    float4 v = ((const float4*)(x + (size_t)c * D_DIM))[lane];
    float* dst = agg + (size_t)r * D_DIM + lane * 4;
    atomicAdd(dst + 0, v.x * norm);
    atomicAdd(dst + 1, v.y * norm);
    atomicAdd(dst + 2, v.z * norm);
    atomicAdd(dst + 3, v.w * norm);
}

// ---------------------------------------------------------------------------
// out = agg @ W^T + b  via native fp32 WMMA (V_WMMA_F32_16X16X4_F32).
// One wave per 16x16 output tile; K=128 fully unrolled in steps of 4.
//
// fp32 A (16x4) fragment layout (ISA 7.12.2): lane l in [0,16) holds
// A[m0+l][k0+0..1] in vgpr0..1; lane l+16 holds A[m0+l][k0+2..3] -> a single
// contiguous float2 load per lane. B (4x16) = W^T, so B-fragment for lane l
// is W[n0+l][k0+2*half .. +1], also one contiguous float2 from row-major W.
// ---------------------------------------------------------------------------
__global__ __launch_bounds__(256) void gcn_gemm_wmma(const float* __restrict__ agg,
                                                     const float* __restrict__ W,
                                                     const float* __restrict__ bias,
                                                     float* __restrict__ out) {
    const int lane = threadIdx.x & 31;
    const int wave = threadIdx.x >> 5;
    const int tile = blockIdx.x * 8 + wave;   // nTilesM * 8 tiles total
    const int mt = tile >> 3;                 // 128 / 16 = 8 N-tiles
    const int nt = tile & 7;
    const int m0 = mt * 16;
    const int n0 = nt * 16;
    const int half = lane >> 4;               // 0: K pair {0,1}, 1: {2,3}
    const int l = lane & 15;

    const float* arow = agg + (size_t)(m0 + l) * D_DIM + 2 * half;
    const float* wrow = W + (size_t)(n0 + l) * D_DIM + 2 * half;

    v8f acc = {};
#pragma unroll
    for (int k0 = 0; k0 < D_DIM; k0 += 4) {
        v2f a = *(const v2f*)(arow + k0);
        v2f b = *(const v2f*)(wrow + k0);
        // 8 args: (neg_a, A, neg_b, B, c_mod, C, reuse_a, reuse_b)
        acc = __builtin_amdgcn_wmma_f32_16x16x4_f32(false, a, false, b,
                                                    (short)0, acc, false, false);
    }

    // D layout: vgpr i -> M = m0 + i + 8*half, N = n0 + l (lanes 0-15 / 16-31)
    float bv = bias[n0 + l];
#pragma unroll
    for (int i = 0; i < 8; ++i) {
        int m = m0 + i + 8 * half;
        out[(size_t)m * D_DIM + n0 + l] = acc[i] + bv;
    }
}

// ---------------------------------------------------------------------------
// Host launcher. Inputs: x [N,128] f32, edge_index [2,E] int32,
// num_nodes (scalar), W [128,128] f32, b [128] f32. Output: [N,128] f32.
// ---------------------------------------------------------------------------
extern "C" void kernel_launch(void* const* d_in, const int* in_sizes, int n_in,
                              void* d_out, int out_size, void* d_ws, size_t ws_size,
                              hipStream_t stream) {
    const float* x  = (const float*)d_in[0];
    const int*   ei = (const int*)d_in[1];
    const float* W  = (const float*)d_in[3];
    const float* b  = (const float*)d_in[4];
    float*       out = (float*)d_out;

    const int nNodes = in_sizes[0] / D_DIM;     // 100000
    const int nEdges = in_sizes[1] / 2;         // 1600000
    const int* row = ei;
    const int* col = ei + nEdges;

    float* agg = (float*)d_ws;                          // nNodes*128 f32
    float* deg = agg + (size_t)nNodes * D_DIM;          // nNodes f32

    // 1) agg = x, deg = 0
    {
        int n4 = nNodes * (D_DIM / 4);
        int nThreads = (n4 > nNodes) ? n4 : nNodes;
        gcn_init<<<(nThreads + 255) / 256, 256, 0, stream>>>(x, agg, deg, n4, nNodes);
    }
    // 2) degree
    gcn_degree<<<(nEdges + 255) / 256, 256, 0, stream>>>(row, deg, nEdges);
    // 3) deg^-0.5
    gcn_rsqrt<<<(nNodes + 255) / 256, 256, 0, stream>>>(deg, nNodes);
    // 4) scatter messages (8 edges per 256-thread block, one wave per edge)
    gcn_scatter<<<(nEdges + 7) / 8, 256, 0, stream>>>(x, row, col, deg, agg, nEdges);
    // 5) WMMA GEMM + bias (100000 % 16 == 0; tiles % 8 == 0 -> EXEC all-1s)
    {
        int nTiles = (nNodes / 16) * (D_DIM / 16);      // 50000
        gcn_gemm_wmma<<<nTiles / 8, 256, 0, stream>>>(agg, W, b, out);
    }
}